// NodeOnlyGlobalModel_21311627722769
// MI455X (gfx1250) — compile-verified
//
#include <hip/hip_runtime.h>
#include <hip/hip_bf16.h>

// ---- problem constants (match reference) ----
#define N_NODES  1048576
#define N_GRAPHS 1024
#define F_X      128
#define F_U      128
#define K_IN     256   // F_X + F_U
#define HIDDEN   512
#define F_OUT    128

typedef __attribute__((ext_vector_type(2))) float v2f;
typedef __attribute__((ext_vector_type(4))) float v4f;
typedef __attribute__((ext_vector_type(8))) float v8f;

// ---------------------------------------------------------------------------
// Kernel 1: segment mean of x over sorted `batch`.
// One block per graph. 256 threads = 8 waves. Lanes cover the 128 features as
// 32 x float4 (512B coalesced per wave iteration); the 8 waves stride over the
// graph's contiguous node range. Non-temporal loads: x (512MB) is read once
// and far exceeds L2, so don't pollute it.
// ---------------------------------------------------------------------------
__global__ __launch_bounds__(256)
void seg_mean_kernel(const float* __restrict__ x,
                     const int*   __restrict__ batch,
                     float*       __restrict__ x_agg) {
    __shared__ v4f part[8][32];

    const int g    = blockIdx.x;
    const int tid  = threadIdx.x;
    const int lane = tid & 31;
    const int grp  = tid >> 5;   // 0..7

    // lower_bound(batch, g) and lower_bound(batch, g+1): batch is sorted.
    int lo = 0, hi = N_NODES;
    while (lo < hi) { int mid = (lo + hi) >> 1; if (batch[mid] <  g) lo = mid + 1; else hi = mid; }
    const int start = lo;
    hi = N_NODES;
    while (lo < hi) { int mid = (lo + hi) >> 1; if (batch[mid] <= g) lo = mid + 1; else hi = mid; }
    const int end = lo;

    v4f acc = (v4f)(0.0f);
    for (int n = start + grp; n < end; n += 8) {
        const v4f* row = reinterpret_cast<const v4f*>(x + (size_t)n * F_X);
        v4f v = __builtin_nontemporal_load(row + lane);
        acc += v;
    }
    part[grp][lane] = acc;
    __syncthreads();

    if (grp == 0) {
        v4f s = part[0][lane];
        #pragma unroll
        for (int i = 1; i < 8; ++i) s += part[i][lane];
        const float cnt = (float)(end - start);
        const float sc  = 1.0f / fmaxf(cnt, 1.0f);
        s *= sc;
        reinterpret_cast<v4f*>(x_agg + (size_t)g * F_X)[lane] = s;
    }
}

// ---------------------------------------------------------------------------
// Kernel 2: fused MLP with V_WMMA_F32_16X16X4_F32.
//   out = relu([x_agg | u] @ W1 + b1) @ W2 + b2
// 64 blocks, 256 threads (8 waves). Block = 16-row M tile.
//   A tile [16,256] staged in LDS (stride padded: K-broadcast reads hit
//   distinct banks). GEMM1 -> h[16,512] in LDS (bias+ReLU). GEMM2 -> out.
// ISA f32 WMMA lane layout:
//   A 16x4 : lane m=lane%16; vgpr0=K(k0+2*hi), vgpr1=K(k0+2*hi+1)
//   B 4x16 : lane n=lane%16; same K split
//   C/D    : vgpr r -> (m = r + 8*hi, n = lane%16)
// ---------------------------------------------------------------------------
__global__ __launch_bounds__(256)
void mlp_wmma_kernel(const float* __restrict__ x_agg,
                     const float* __restrict__ u,
                     const float* __restrict__ W1,
                     const float* __restrict__ b1,
                     const float* __restrict__ W2,
                     const float* __restrict__ b2,
                     float*       __restrict__ out) {
    constexpr int AS = K_IN  + 4;   // padded LDS strides (bank-conflict free
    constexpr int HS = HIDDEN + 4;  // for stride-AS/HS column walks)
    __shared__ float Atile[16 * AS];
    __shared__ float Htile[16 * HS];

    const int tid = threadIdx.x;
    const int m0  = blockIdx.x * 16;

    // Stage concat tile [16 x 256] = [x_agg rows | u rows]
    for (int i = tid; i < 16 * K_IN; i += 256) {
        const int m = i >> 8;         // /256
        const int c = i & (K_IN - 1); // %256
        const float v = (c < F_X) ? x_agg[(size_t)(m0 + m) * F_X + c]
                                  : u[(size_t)(m0 + m) * F_U + (c - F_X)];
        Atile[m * AS + c] = v;
    }
    __syncthreads();

    const int wave = tid >> 5;
    const int lane = tid & 31;
    const int lm   = lane & 15;
    const int hi   = lane >> 4;      // 0: K pair {k0,k0+1}; 1: {k0+2,k0+3}

    // ---- GEMM1: [16,256] @ [256,512] -> relu -> Htile ----
    for (int nt = wave; nt < HIDDEN / 16; nt += 8) {
        const int n0 = nt * 16;
        v8f acc = {};
        for (int k0 = 0; k0 < K_IN; k0 += 4) {
            const int ka = k0 + 2 * hi;
            v2f a; a[0] = Atile[lm * AS + ka];
                   a[1] = Atile[lm * AS + ka + 1];
            v2f b; b[0] = W1[(size_t)ka       * HIDDEN + n0 + lm];
                   b[1] = W1[(size_t)(ka + 1) * HIDDEN + n0 + lm];
            acc = __builtin_amdgcn_wmma_f32_16x16x4_f32(
                false, a, false, b, (short)0, acc, false, false);
        }
        const float bias = b1[n0 + lm];
        #pragma unroll
        for (int r = 0; r < 8; ++r) {
            const float v = acc[r] + bias;
            Htile[(r + 8 * hi) * HS + n0 + lm] = fmaxf(v, 0.0f);
        }
    }
    __syncthreads();

    // ---- GEMM2: [16,512] @ [512,128] + b2 -> out ----
    {
        const int n0 = wave * 16;    // 8 waves x 16 = 128 output cols
        v8f acc = {};
        for (int k0 = 0; k0 < HIDDEN; k0 += 4) {
            const int ka = k0 + 2 * hi;
            v2f a; a[0] = Htile[lm * HS + ka];
                   a[1] = Htile[lm * HS + ka + 1];
            v2f b; b[0] = W2[(size_t)ka       * F_OUT + n0 + lm];
                   b[1] = W2[(size_t)(ka + 1) * F_OUT + n0 + lm];
            acc = __builtin_amdgcn_wmma_f32_16x16x4_f32(
                false, a, false, b, (short)0, acc, false, false);
        }
        const float bias = b2[n0 + lm];
        #pragma unroll
        for (int r = 0; r < 8; ++r) {
            out[(size_t)(m0 + r + 8 * hi) * F_OUT + n0 + lm] = acc[r] + bias;
        }
    }
}

extern "C" void kernel_launch(void* const* d_in, const int* in_sizes, int n_in,
                              void* d_out, int out_size, void* d_ws, size_t ws_size,
                              hipStream_t stream) {
    // setup_inputs order: x, edge_index, edge_attr, u, batch, W1, b1, W2, b2
    const float* x     = (const float*)d_in[0];
    // d_in[1] edge_index, d_in[2] edge_attr: unused by the reference math
    const float* u     = (const float*)d_in[3];
    const int*   batch = (const int*)d_in[4];
    const float* W1    = (const float*)d_in[5];
    const float* b1    = (const float*)d_in[6];
    const float* W2    = (const float*)d_in[7];
    const float* b2    = (const float*)d_in[8];
    float* out = (float*)d_out;

    float* x_agg = (float*)d_ws;   // [N_GRAPHS, F_X] f32 = 512 KB scratch

    seg_mean_kernel<<<N_GRAPHS, 256, 0, stream>>>(x, batch, x_agg);
    mlp_wmma_kernel<<<N_GRAPHS / 16, 256, 0, stream>>>(x_agg, u, W1, b1, W2, b2, out);
}